// GCN_WWW_35871566856584
// MI455X (gfx1250) — compile-verified
//
#include <hip/hip_runtime.h>
#include <hip/hip_bf16.h>

typedef float v2f __attribute__((ext_vector_type(2)));
typedef float v8f __attribute__((ext_vector_type(8)));

// ---------------- zero fill (float4) ----------------
__global__ void fill_zero_f4(float4* __restrict__ p, long long n4) {
  long long i = (long long)blockIdx.x * blockDim.x + threadIdx.x;
  if (i < n4) p[i] = make_float4(0.f, 0.f, 0.f, 0.f);
}

// ---------------- SpMM: out[rows[e],:] += vals[e] * dense[cols[e],:] ----------------
// 2^lc float4-chunks per edge (lc = log2(D/4)). Dense operand is L2-resident (<=102MB
// vs 192MB L2), so gathers+atomics run at L2 speed, not HBM.
__global__ void spmm_atomic(const int* __restrict__ rows, const int* __restrict__ cols,
                            const float* __restrict__ vals, const float* __restrict__ dense,
                            float* __restrict__ out, long long E, int lc, int D) {
  long long tid = (long long)blockIdx.x * blockDim.x + threadIdx.x;
  long long total = E << lc;
  if (tid >= total) return;
  long long e = tid >> lc;
  int c4 = (int)(tid & ((1LL << lc) - 1));
  int r = rows[e];
  int c = cols[e];
  float v = vals[e];
  const float4* dp = (const float4*)(dense + (long long)c * D);
  float4 d = dp[c4];
  float* o = out + (long long)r * D + 4LL * c4;
  atomicAdd(o + 0, v * d.x);
  atomicAdd(o + 1, v * d.y);
  atomicAdd(o + 2, v * d.z);
  atomicAdd(o + 3, v * d.w);
}

// ---------------- bias (+ optional relu), in place; D is a power of two ----------------
template <bool RELU>
__global__ void bias_act(float* __restrict__ y, const float* __restrict__ b,
                         long long n, int dmask) {
  long long i = (long long)blockIdx.x * blockDim.x + threadIdx.x;
  if (i >= n) return;
  float t = y[i] + b[i & dmask];
  if (RELU) t = fmaxf(t, 0.f);
  y[i] = t;
}

// =============================================================================
// LDS-staged WMMA fp32 GEMM: C[M,Dout] = A[M,K] @ W[K,Dout]
// Block = 256 threads (8 waves). Block tile = 128(M) x 64(N).
// W slab staged through LDS in K-chunks of KC=64, double-buffered (2 x 16KB).
// Inner loop: ds_load_b64 (B frag) + v_wmma_f32_16x16x4_f32 only; A fragments
// for the whole chunk preloaded to registers; next chunk's W prefetched to
// registers during compute. M%16==0, K%KC==0, Dout%64==0.
// =============================================================================
#define KC 64
#define BN 64

__global__ void gemm_wmma_lds(const float* __restrict__ A, const float* __restrict__ W,
                              float* __restrict__ C, int M, int K, int Dout) {
  // Swizzled: ldsB[buf][kpair][n][k&1] -> B frag (k,k+1 at column n) is one b64.
  __shared__ float ldsB[2][KC / 2][BN][2];  // 2 x 16 KB

  const int tid = threadIdx.x;
  const int wave = tid >> 5;
  const int lane = tid & 31;

  const int nblocks = Dout / BN;
  const int bm = (int)blockIdx.x / nblocks;
  const int bn = (int)blockIdx.x % nblocks;
  const int n0 = bn * BN;

  const int tm = bm * 8 + wave;            // this wave's 16-row tile
  const bool active = (tm * 16 < M);
  const int m0 = active ? tm * 16 : 0;     // clamp: inactive waves compute garbage, skip store
  const int mr = lane & 15;                // A row / C column within tile
  const int kh = (lane >> 4) * 2;          // 16x4 f32 A/B layout: upper half-lanes own K+2,K+3

  const int nchunks = K / KC;
  const float* arow = A + (long long)(m0 + mr) * K;

  v8f acc[4];
#pragma unroll
  for (int t = 0; t < 4; ++t)
#pragma unroll
    for (int r = 0; r < 8; ++r) acc[t][r] = 0.f;

  // Staging registers: 8 float2 per thread per chunk (2048 pairs / 256 threads).
  v2f sreg[8];
#pragma unroll
  for (int it = 0; it < 8; ++it) {
    int idx = it * 256 + tid;
    int p = idx >> 6;          // k-pair within chunk
    int n = idx & 63;
    const float* wp = W + (long long)(2 * p) * Dout + n0 + n;
    sreg[it].x = wp[0];
    sreg[it].y = wp[Dout];
  }

  for (int kc = 0; kc < nchunks; ++kc) {
    const int buf = kc & 1;
    const int kb = kc * KC;

    // Commit staged registers to LDS (conflict-free b64 stores).
#pragma unroll
    for (int it = 0; it < 8; ++it) {
      int idx = it * 256 + tid;
      int p = idx >> 6;
      int n = idx & 63;
      *(v2f*)&ldsB[buf][p][n][0] = sreg[it];
    }
    __syncthreads();

    // Prefetch next chunk's W slab into registers (overlaps the WMMA compute).
    if (kc + 1 < nchunks) {
      const int kb2 = kb + KC;
#pragma unroll
      for (int it = 0; it < 8; ++it) {
        int idx = it * 256 + tid;
        int p = idx >> 6;
        int n = idx & 63;
        const float* wp = W + (long long)(kb2 + 2 * p) * Dout + n0 + n;
        sreg[it].x = wp[0];
        sreg[it].y = wp[Dout];
      }
    }

    // Preload all A fragments for this chunk (streaming b64 loads, one clause).
    v2f afrag[KC / 4];
#pragma unroll
    for (int s = 0; s < KC / 4; ++s) {
      const float* ap = arow + kb + s * 4 + kh;
      afrag[s].x = ap[0];
      afrag[s].y = ap[1];
    }

    // Compute: pure ds_load_b64 + wmma.
#pragma unroll
    for (int s = 0; s < KC / 4; ++s) {
      const int pair = s * 2 + (kh >> 1);
#pragma unroll
      for (int t = 0; t < 4; ++t) {
        v2f b = *(const v2f*)&ldsB[buf][pair][t * 16 + mr][0];
        acc[t] = __builtin_amdgcn_wmma_f32_16x16x4_f32(
            false, afrag[s], false, b, (short)0, acc[t], false, false);
      }
    }
    __syncthreads();  // buffer `buf` is reused at chunk kc+2; this barrier protects it
  }

  if (active) {
    // C/D layout: VGPR r -> row m0+r (lanes 0-15) / m0+r+8 (lanes 16-31), col = lane&15
    const int mbase = m0 + (lane >> 4) * 8;
#pragma unroll
    for (int t = 0; t < 4; ++t) {
      float* crow = C + (long long)mbase * Dout + n0 + t * 16 + mr;
#pragma unroll
      for (int r = 0; r < 8; ++r) crow[(long long)r * Dout] = acc[t][r];
    }
  }
}

// ---------------- small WMMA GEMM for the final layer: relu(A)@Wd + bd ----------------
// One wave = one 16x16 output tile; K=128, Dout=16. ReLU fused into A-load,
// bias fused into accumulator init.
__global__ void gemm_wmma_out(const float* __restrict__ A, const float* __restrict__ W,
                              const float* __restrict__ bias, float* __restrict__ C,
                              int M, int K, int Dout) {
  int wid = ((int)blockIdx.x * (int)blockDim.x + (int)threadIdx.x) >> 5;
  int lane = threadIdx.x & 31;
  if (wid * 16 >= M) return;  // wave-uniform
  int m0 = wid * 16;
  int mr = lane & 15;
  int kh = (lane >> 4) * 2;

  v8f acc;
  float bv = bias[mr];
#pragma unroll
  for (int r = 0; r < 8; ++r) acc[r] = bv;

  const float* arow = A + (long long)(m0 + mr) * K;
#pragma unroll 4
  for (int k = 0; k < K; k += 4) {
    v2f a;
    a.x = fmaxf(arow[k + kh], 0.f);
    a.y = fmaxf(arow[k + kh + 1], 0.f);
    const float* wr = W + (long long)(k + kh) * Dout + mr;
    v2f b;
    b.x = wr[0];
    b.y = wr[Dout];
    acc = __builtin_amdgcn_wmma_f32_16x16x4_f32(
        false, a, false, b, (short)0, acc, false, false);
  }

  int mbase = m0 + (lane >> 4) * 8;
  float* crow = C + (long long)mbase * Dout + mr;
#pragma unroll
  for (int r = 0; r < 8; ++r) crow[(long long)r * Dout] = acc[r];
}

// ---------------- host-side launch helpers ----------------
static inline int cdiv_ll(long long a, int b) { return (int)((a + b - 1) / b); }

static inline void launch_zero(float* p, long long n, hipStream_t s) {
  long long n4 = n >> 2;
  hipLaunchKernelGGL(fill_zero_f4, dim3(cdiv_ll(n4, 256)), dim3(256), 0, s,
                     (float4*)p, n4);
}

static inline void launch_spmm(const int* rows, const int* cols, const float* vals,
                               const float* dense, float* out, long long E, int D,
                               hipStream_t s) {
  int lc = (D == 256) ? 6 : 5;  // log2(D/4)
  long long total = E << lc;
  hipLaunchKernelGGL(spmm_atomic, dim3(cdiv_ll(total, 256)), dim3(256), 0, s,
                     rows, cols, vals, dense, out, E, lc, D);
}

static inline void launch_gemm_lds(const float* A, const float* W, float* C,
                                   int M, int K, int Dout, hipStream_t s) {
  int mblocks = (M / 16 + 7) / 8;
  int nblocks = Dout / BN;
  hipLaunchKernelGGL(gemm_wmma_lds, dim3(mblocks * nblocks), dim3(256), 0, s,
                     A, W, C, M, K, Dout);
}

extern "C" void kernel_launch(void* const* d_in, const int* in_sizes, int n_in,
                              void* d_out, int out_size, void* d_ws, size_t ws_size,
                              hipStream_t stream) {
  const float* x  = (const float*)d_in[0];
  const float* W1 = (const float*)d_in[16];
  const float* b1 = (const float*)d_in[17];
  const float* W2 = (const float*)d_in[18];
  const float* b2 = (const float*)d_in[19];
  const float* W3 = (const float*)d_in[20];
  const float* b3 = (const float*)d_in[21];
  const float* Wd = (const float*)d_in[22];
  const float* bd = (const float*)d_in[23];

  const int       N = in_sizes[0] / 512;   // 100000, divisible by 16
  const long long E = in_sizes[1];         // 3.2M

  float* out = (float*)d_out;
  float* XW1 = (float*)d_ws;                       // N*256 (persistent across branches)
  float* B1  = XW1 + (long long)N * 256;           // N*256 ping
  float* B2  = B1 + (long long)N * 256;            // N*256 pong

  // Shared across all branches: XW1 = x @ W1  (K=512, Dout=256)
  launch_gemm_lds(x, W1, XW1, N, 512, 256, stream);

  // Output order: x1<-adj5, x2<-adj4, x3<-adj3, x4<-adj1 (adj2 unused).
  const int bases[4] = {13, 10, 7, 1};
  for (int br = 0; br < 4; ++br) {
    const int*   rows = (const int*)d_in[bases[br]];
    const int*   cols = (const int*)d_in[bases[br] + 1];
    const float* vals = (const float*)d_in[bases[br] + 2];
    float*       ob   = out + (long long)br * N * 16;

    // h1 = relu(spmm(A, XW1) + b1) -> B1
    launch_zero(B1, (long long)N * 256, stream);
    launch_spmm(rows, cols, vals, XW1, B1, E, 256, stream);
    hipLaunchKernelGGL((bias_act<true>), dim3(cdiv_ll((long long)N * 256, 256)),
                       dim3(256), 0, stream, B1, b1, (long long)N * 256, 255);

    // t2 = h1 @ W2 (K=256, Dout=256) -> B2
    launch_gemm_lds(B1, W2, B2, N, 256, 256, stream);

    // h2 = relu(spmm(A, t2) + b2) -> B1
    launch_zero(B1, (long long)N * 256, stream);
    launch_spmm(rows, cols, vals, B2, B1, E, 256, stream);
    hipLaunchKernelGGL((bias_act<true>), dim3(cdiv_ll((long long)N * 256, 256)),
                       dim3(256), 0, stream, B1, b2, (long long)N * 256, 255);

    // t3 = h2 @ W3 (K=256, Dout=128) -> B2
    launch_gemm_lds(B1, W3, B2, N, 256, 128, stream);

    // h3 = spmm(A, t3) + b3 (no relu) -> B1
    launch_zero(B1, (long long)N * 128, stream);
    launch_spmm(rows, cols, vals, B2, B1, E, 128, stream);
    hipLaunchKernelGGL((bias_act<false>), dim3(cdiv_ll((long long)N * 128, 256)),
                       dim3(256), 0, stream, B1, b3, (long long)N * 128, 127);

    // out_b = relu(h3) @ Wd + bd (K=128, Dout=16) -> d_out slice
    {
      int waves = N / 16;
      hipLaunchKernelGGL(gemm_wmma_out, dim3(cdiv_ll((long long)waves * 32, 256)),
                         dim3(256), 0, stream, B1, Wd, bd, ob, N, 128, 16);
    }
  }
}